// MGDPR_25709674234638
// MI455X (gfx1250) — compile-verified
//
#include <hip/hip_runtime.h>

typedef __attribute__((ext_vector_type(2))) float v2f;
typedef __attribute__((ext_vector_type(8))) float v8f;

#define BB 4
#define NN 512
#define TT 64
#define FD 8
#define DD 128
#define LL 2
#define HH 4
#define HDm 32
#define BN 2048
#define LDST 132   // padded LDS row stride for [64][128] tiles (132 mod 64 banks = 4)
#define LDSA 68    // padded stride for attn [64][64]
#define EPSf 1e-5f
#define LRELUf 0.2f

__device__ __forceinline__ v8f wmma_f32(v2f a, v2f b, v8f c) {
  // D = A(16x4 f32) * B(4x16 f32) + C(16x16 f32)
  return __builtin_amdgcn_wmma_f32_16x16x4_f32(false, a, false, b, (short)0, c, false, false);
}
__device__ __forceinline__ float lrelu(float x) { return x > 0.f ? x : LRELUf * x; }

// ---------------- prep kernels ----------------

__global__ void transpose_kernel(const float* __restrict__ src, float* __restrict__ dst,
                                 int R, int C) {
  int i = blockIdx.x * blockDim.x + threadIdx.x;
  if (i < R * C) {
    int r = i / C, c = i % C;
    dst[(size_t)c * R + r] = src[i];
  }
}

__global__ void rowsum_kernel(const float* __restrict__ adj, float* __restrict__ rs) {
  __shared__ float p[128];
  int row = blockIdx.x, t = threadIdx.x;
  const float* a = adj + (size_t)row * NN;
  float s = 0.f;
  for (int j = t; j < NN; j += 128) s += a[j];
  p[t] = s;
  __syncthreads();
  for (int off = 64; off > 0; off >>= 1) {
    if (t < off) p[t] += p[t + off];
    __syncthreads();
  }
  if (t == 0) rs[row] = p[0] + 1e-9f;
}

__global__ void proj_kernel(const float* __restrict__ x, const float* __restrict__ pw,
                            const float* __restrict__ pb, const float* __restrict__ pos,
                            float* __restrict__ h) {
  int i = blockIdx.x * blockDim.x + threadIdx.x;  // exact grid: BN*TT*DD
  int d = i & 127;
  int t = (i >> 7) & 63;
  int bn = i >> 13;
  const float* xr = x + ((size_t)bn * TT + t) * FD;
  const float* wr = pw + d * FD;
  float s = pb[d] + pos[t * DD + d];
#pragma unroll
  for (int f = 0; f < FD; ++f) s += xr[f] * wr[f];
  h[i] = s;
}

// ---------------- fused retention layer ----------------
// One workgroup (8 wave32) per (b,n) sample: QKV GEMMs, per-head retention,
// group-norm, layer-norm; everything staged in LDS.

__global__ __launch_bounds__(256) void retention_kernel(
    const float* __restrict__ h,
    const float* __restrict__ wqT, const float* __restrict__ wkT, const float* __restrict__ wvT,
    const float* __restrict__ qb, const float* __restrict__ kbv, const float* __restrict__ vb,
    const float* __restrict__ gnw, const float* __restrict__ gnb,
    const float* __restrict__ lnw, const float* __restrict__ lnb,
    const float* __restrict__ gam,
    float* __restrict__ h_ret) {
  __shared__ float hs[TT * LDST];   // h tile, later reused as retention output
  __shared__ float Qs[TT * LDST];
  __shared__ float Ks[TT * LDST];
  __shared__ float Vs[TT * LDST];
  __shared__ float attn_s[TT * LDSA];
  __shared__ float red[256], red2[256];
  __shared__ float gstats[HH * 2];
  __shared__ float rowst[TT * 2];

  const int tid = threadIdx.x;
  const int lane = tid & 31;
  const int wave = tid >> 5;
  const int bn = blockIdx.x;
  const float* hg = h + (size_t)bn * TT * DD;

  // load h tile (coalesced float4)
  for (int i = tid; i < TT * DD / 4; i += 256) {
    int e = i << 2;
    int r = e >> 7, c = e & 127;
    *(float4*)&hs[r * LDST + c] = *(const float4*)(hg + e);
  }
  __syncthreads();

  const int m15 = lane & 15;
  const int koff = (lane >> 4) << 1;
  const int crh = (lane >> 4) << 3;

  // ---- Q/K/V = h @ W^T + b : 3 GEMMs [64,128]x[128,128], 96 tiles over 8 waves
  for (int it = 0; it < 12; ++it) {
    int tileId = wave + (it << 3);
    int g = tileId >> 5;
    int w32 = tileId & 31;
    int mt = w32 >> 3, nt = w32 & 7;
    const float* wT = (g == 0) ? wqT : (g == 1) ? wkT : wvT;
    const float* bias = (g == 0) ? qb : (g == 1) ? kbv : vb;
    float* outp = (g == 0) ? Qs : (g == 1) ? Ks : Vs;
    int arow = (mt << 4) + m15;
    int ncol = (nt << 4) + m15;
    v8f acc = {};
    for (int k0 = 0; k0 < DD; k0 += 4) {
      const float* ap = &hs[arow * LDST + k0 + koff];
      v2f a; a[0] = ap[0]; a[1] = ap[1];
      const float* bp = wT + (size_t)(k0 + koff) * DD + ncol;
      v2f b; b[0] = bp[0]; b[1] = bp[DD];
      acc = wmma_f32(a, b, acc);
    }
    int cr = (mt << 4) + crh;
    float bvv = bias[ncol];
#pragma unroll
    for (int v = 0; v < 8; ++v) outp[(cr + v) * LDST + ncol] = acc[v] + bvv;
  }
  __syncthreads();

  // ---- per-head retention: attn = (Q K^T) * decay ; out = attn @ V
  const float lg = __logf(gam[0]);
  for (int head = 0; head < HH; ++head) {
    const int cb = head * HDm;
    // attn tiles: 4x4 of 16x16, 2 per wave
    for (int j = 0; j < 2; ++j) {
      int t16 = (wave << 1) + j;
      int qt = t16 >> 2, kt = t16 & 3;
      int arow = (qt << 4) + m15;
      int bcol = (kt << 4) + m15;  // time index on K side
      v8f acc = {};
      for (int k0 = 0; k0 < HDm; k0 += 4) {
        const float* ap = &Qs[arow * LDST + cb + k0 + koff];
        v2f a; a[0] = ap[0]; a[1] = ap[1];
        const float* bp = &Ks[bcol * LDST + cb + k0 + koff];  // B[k][n] = K[n][k]
        v2f b; b[0] = bp[0]; b[1] = bp[1];
        acc = wmma_f32(a, b, acc);
      }
      int cr = (qt << 4) + crh;
      int cc = (kt << 4) + m15;
#pragma unroll
      for (int v = 0; v < 8; ++v) {
        int q = cr + v;
        float val = (q >= cc) ? acc[v] * __expf((float)(q - cc) * lg) : 0.f;
        attn_s[q * LDSA + cc] = val;
      }
    }
    __syncthreads();
    // out tiles: [64 x 32] -> 8 tiles, 1 per wave; reuse hs as output buffer
    {
      int mt = wave >> 1, nt = wave & 1;
      int arow = (mt << 4) + m15;
      int ncol = (nt << 4) + m15;
      v8f acc = {};
      for (int k0 = 0; k0 < TT; k0 += 4) {
        const float* ap = &attn_s[arow * LDSA + k0 + koff];
        v2f a; a[0] = ap[0]; a[1] = ap[1];
        const float* bp = &Vs[(k0 + koff) * LDST + cb + ncol];  // B[k][n] = V[k][n]
        v2f b; b[0] = bp[0]; b[1] = bp[LDST];
        acc = wmma_f32(a, b, acc);
      }
      int cr = (mt << 4) + crh;
#pragma unroll
      for (int v = 0; v < 8; ++v) hs[(cr + v) * LDST + cb + ncol] = acc[v];
    }
    __syncthreads();
  }

  // ---- group norm stats: per head over (T=64, hd=32) = 2048 elems
  {
    int head = tid >> 6, idx = tid & 63;
    float s = 0.f, sq = 0.f;
    for (int j = 0; j < 32; ++j) {
      int e = idx + (j << 6);
      int r = e >> 5, c = (e & 31) + head * HDm;
      float v = hs[r * LDST + c];
      s += v; sq += v * v;
    }
    red[tid] = s; red2[tid] = sq;
    __syncthreads();
    for (int off = 32; off > 0; off >>= 1) {
      if (idx < off) { red[tid] += red[tid + off]; red2[tid] += red2[tid + off]; }
      __syncthreads();
    }
    if (idx == 0) {
      float mean = red[tid] * (1.f / 2048.f);
      float var = red2[tid] * (1.f / 2048.f) - mean * mean;
      gstats[head * 2] = mean;
      gstats[head * 2 + 1] = rsqrtf(var + EPSf);
    }
    __syncthreads();
  }
  // ---- apply GN + per-row LN stats (4 threads per row, 32 cols each; col chunk == head)
  {
    int r = tid >> 2, p = tid & 3;
    float m = gstats[p * 2], rsd = gstats[p * 2 + 1];
    float s = 0.f, sq = 0.f;
    for (int j = 0; j < 32; ++j) {
      int c = (p << 5) + j;
      float v = (hs[r * LDST + c] - m) * rsd * gnw[c] + gnb[c];
      hs[r * LDST + c] = v;
      s += v; sq += v * v;
    }
    red[tid] = s; red2[tid] = sq;
    __syncthreads();
    if (p == 0) {
      float S = red[tid] + red[tid + 1] + red[tid + 2] + red[tid + 3];
      float Q2 = red2[tid] + red2[tid + 1] + red2[tid + 2] + red2[tid + 3];
      float mu = S * (1.f / 128.f);
      float var = Q2 * (1.f / 128.f) - mu * mu;
      rowst[r * 2] = mu;
      rowst[r * 2 + 1] = rsqrtf(var + EPSf);
    }
    __syncthreads();
  }
  // ---- LN apply + store h_ret
  {
    float* hrg = h_ret + (size_t)bn * TT * DD;
    for (int i = tid; i < TT * DD; i += 256) {
      int r = i >> 7, c = i & 127;
      hrg[i] = (hs[r * LDST + c] - rowst[r * 2]) * rowst[r * 2 + 1] * lnw[c] + lnb[c];
    }
  }
}

// ---------------- diffusion ----------------
// cur0 = h_last @ diff_w^T ; acc = w0 * cur0.  M=2048, K=128, N=128; 1 tile per wave.
__global__ __launch_bounds__(256) void diff_init_kernel(
    const float* __restrict__ h_ret, const float* __restrict__ dfT,
    const float* __restrict__ cw, float* __restrict__ cur, float* __restrict__ acc_out) {
  const int tid = threadIdx.x, lane = tid & 31, wave = tid >> 5;
  int tile = blockIdx.x * 8 + wave;  // 0..1023
  int mt = tile >> 3, nt = tile & 7;
  const int m15 = lane & 15, koff = (lane >> 4) << 1;
  int arow = (mt << 4) + m15;
  int ncol = (nt << 4) + m15;
  const float* ar = h_ret + ((size_t)arow * TT + (TT - 1)) * DD;
  v8f acc = {};
  for (int k0 = 0; k0 < DD; k0 += 4) {
    v2f a; a[0] = ar[k0 + koff]; a[1] = ar[k0 + koff + 1];
    const float* bp = dfT + (size_t)(k0 + koff) * DD + ncol;
    v2f b; b[0] = bp[0]; b[1] = bp[DD];
    acc = wmma_f32(a, b, acc);
  }
  int cr = (mt << 4) + ((lane >> 4) << 3);
  float w0 = cw[0];
#pragma unroll
  for (int v = 0; v < 8; ++v) {
    size_t o = (size_t)(cr + v) * DD + ncol;
    cur[o] = acc[v];
    acc_out[o] = w0 * acc[v];
  }
}

// cur_out = (adj[b] @ cur_in[b]) / rowsum ; acc += cw[kidx]*cur_out.  Per b: M=512,K=512,N=128.
__global__ __launch_bounds__(256) void diff_hop_kernel(
    const float* __restrict__ adj, const float* __restrict__ cur_in,
    const float* __restrict__ rs, const float* __restrict__ cw, int kidx,
    float* __restrict__ cur_out, float* __restrict__ acc_io) {
  const int tid = threadIdx.x, lane = tid & 31, wave = tid >> 5;
  int tile = blockIdx.x * 8 + wave;  // 0..1023
  int b = tile >> 8;
  int w = tile & 255;
  int mt = w >> 3, nt = w & 7;
  const int m15 = lane & 15, koff = (lane >> 4) << 1;
  int lrow = (mt << 4) + m15;
  int ncol = (nt << 4) + m15;
  const float* adjr = adj + ((size_t)b * NN + lrow) * NN;
  const float* curb = cur_in + (size_t)b * NN * DD;
  v8f acc = {};
  for (int k0 = 0; k0 < NN; k0 += 4) {
    v2f a; a[0] = adjr[k0 + koff]; a[1] = adjr[k0 + koff + 1];
    const float* bp = curb + (size_t)(k0 + koff) * DD + ncol;
    v2f b2; b2[0] = bp[0]; b2[1] = bp[DD];
    acc = wmma_f32(a, b2, acc);
  }
  int cr = (mt << 4) + ((lane >> 4) << 3);
  float wk = cw[kidx];
#pragma unroll
  for (int v = 0; v < 8; ++v) {
    int grow = b * NN + cr + v;
    float val = acc[v] / rs[grow];
    size_t o = (size_t)grow * DD + ncol;
    cur_out[o] = val;
    acc_io[o] += wk * val;
  }
}

__global__ void diff_act_kernel(float* __restrict__ acc, const float* __restrict__ cb) {
  int i = blockIdx.x * blockDim.x + threadIdx.x;  // exact grid BN*DD
  acc[i] = lrelu(acc[i] + cb[0]);
}

// ---------------- concat-MLP fusion ----------------
// h = lrelu([h_ret, h_diff] @ mlp^T + b).  The h_diff half is row-invariant per (b,n):
// fold it into a per-(b,n) 128-vector, then one WMMA GEMM for the h_ret half.
__global__ __launch_bounds__(256) void fusion_kernel(
    const float* __restrict__ h_ret, const float* __restrict__ h_diff,
    const float* __restrict__ mlpT, const float* __restrict__ mb,
    float* __restrict__ h_out) {
  __shared__ float hr[TT * LDST];
  __shared__ float hd[DD];
  __shared__ float v2s[DD];
  const int tid = threadIdx.x, lane = tid & 31, wave = tid >> 5;
  const int bn = blockIdx.x;
  const float* hrg = h_ret + (size_t)bn * TT * DD;
  for (int i = tid; i < TT * DD / 4; i += 256) {
    int e = i << 2;
    int r = e >> 7, c = e & 127;
    *(float4*)&hr[r * LDST + c] = *(const float4*)(hrg + e);
  }
  if (tid < DD) hd[tid] = h_diff[(size_t)bn * DD + tid];
  __syncthreads();
  if (tid < DD) {
    float s = mb[tid];
    for (int k = 0; k < DD; ++k) s += hd[k] * mlpT[(size_t)(DD + k) * DD + tid];
    v2s[tid] = s;
  }
  __syncthreads();
  const int m15 = lane & 15, koff = (lane >> 4) << 1;
  float* hog = h_out + (size_t)bn * TT * DD;
  for (int it = 0; it < 4; ++it) {
    int tile = wave + (it << 3);  // 0..31
    int mt = tile >> 3, nt = tile & 7;
    int arow = (mt << 4) + m15;
    int ncol = (nt << 4) + m15;
    v8f acc = {};
    for (int k0 = 0; k0 < DD; k0 += 4) {
      const float* ap = &hr[arow * LDST + k0 + koff];
      v2f a; a[0] = ap[0]; a[1] = ap[1];
      const float* bp = mlpT + (size_t)(k0 + koff) * DD + ncol;
      v2f b; b[0] = bp[0]; b[1] = bp[DD];
      acc = wmma_f32(a, b, acc);
    }
    int cr = (mt << 4) + ((lane >> 4) << 3);
    float add = v2s[ncol];
#pragma unroll
    for (int v = 0; v < 8; ++v) hog[(size_t)(cr + v) * DD + ncol] = lrelu(acc[v] + add);
  }
}

// ---------------- output head ----------------
__global__ __launch_bounds__(64) void head_kernel(
    const float* __restrict__ h, const float* __restrict__ w1, const float* __restrict__ b1,
    const float* __restrict__ w2, const float* __restrict__ b2, float* __restrict__ out) {
  __shared__ float last[DD];
  __shared__ float red[64];
  int row = blockIdx.x, t = threadIdx.x;
  const float* lg = h + ((size_t)row * TT + (TT - 1)) * DD;
  last[t] = lg[t];
  last[t + 64] = lg[t + 64];
  __syncthreads();
  float s = b1[t];
  const float* wr = w1 + (size_t)t * DD;
  for (int k = 0; k < DD; ++k) s += last[k] * wr[k];
  red[t] = lrelu(s) * w2[t];
  __syncthreads();
  for (int off = 32; off > 0; off >>= 1) {
    if (t < off) red[t] += red[t + off];
    __syncthreads();
  }
  if (t == 0) out[row] = red[0] + b2[0];
}

// ---------------- host orchestration ----------------
extern "C" void kernel_launch(void* const* d_in, const int* in_sizes, int n_in,
                              void* d_out, int out_size, void* d_ws, size_t ws_size,
                              hipStream_t stream) {
  const float* x = (const float*)d_in[0];
  const float* adj = (const float*)d_in[1];
  const float* proj_w = (const float*)d_in[2];
  const float* proj_b = (const float*)d_in[3];
  const float* pos = (const float*)d_in[4];
  const float* diff_w = (const float*)d_in[5];
  const float* conv_w = (const float*)d_in[6];
  const float* conv_b = (const float*)d_in[7];
  const float* wq_w = (const float*)d_in[8];
  const float* wq_b = (const float*)d_in[9];
  const float* wk_w = (const float*)d_in[10];
  const float* wk_b = (const float*)d_in[11];
  const float* wv_w = (const float*)d_in[12];
  const float* wv_b = (const float*)d_in[13];
  const float* gn_w = (const float*)d_in[14];
  const float* gn_b = (const float*)d_in[15];
  const float* gamma = (const float*)d_in[16];
  const float* ln_w = (const float*)d_in[17];
  const float* ln_b = (const float*)d_in[18];
  const float* mlp_w = (const float*)d_in[19];
  const float* mlp_b = (const float*)d_in[20];
  const float* head1_w = (const float*)d_in[21];
  const float* head1_b = (const float*)d_in[22];
  const float* head2_w = (const float*)d_in[23];
  const float* head2_b = (const float*)d_in[24];
  float* out = (float*)d_out;

  float* ws = (float*)d_ws;
  size_t o = 0;
  float* h = ws + o;    o += (size_t)BN * TT * DD;
  float* hret = ws + o; o += (size_t)BN * TT * DD;
  float* acc = ws + o;  o += (size_t)BN * DD;
  float* cura = ws + o; o += (size_t)BN * DD;
  float* curb = ws + o; o += (size_t)BN * DD;
  float* rsum = ws + o; o += BN;
  float* wqT = ws + o;  o += (size_t)LL * DD * DD;
  float* wkT = ws + o;  o += (size_t)LL * DD * DD;
  float* wvT = ws + o;  o += (size_t)LL * DD * DD;
  float* dfT = ws + o;  o += (size_t)LL * DD * DD;
  float* mlpT = ws + o; o += (size_t)LL * 2 * DD * DD;

  const int TB = 256;
  for (int l = 0; l < LL; ++l) {
    transpose_kernel<<<(DD * DD + TB - 1) / TB, TB, 0, stream>>>(wq_w + l * DD * DD, wqT + l * DD * DD, DD, DD);
    transpose_kernel<<<(DD * DD + TB - 1) / TB, TB, 0, stream>>>(wk_w + l * DD * DD, wkT + l * DD * DD, DD, DD);
    transpose_kernel<<<(DD * DD + TB - 1) / TB, TB, 0, stream>>>(wv_w + l * DD * DD, wvT + l * DD * DD, DD, DD);
    transpose_kernel<<<(DD * DD + TB - 1) / TB, TB, 0, stream>>>(diff_w + l * DD * DD, dfT + l * DD * DD, DD, DD);
    transpose_kernel<<<(2 * DD * DD + TB - 1) / TB, TB, 0, stream>>>(mlp_w + l * 2 * DD * DD, mlpT + l * 2 * DD * DD, DD, 2 * DD);
  }
  rowsum_kernel<<<BN, 128, 0, stream>>>(adj, rsum);
  proj_kernel<<<(BN * TT * DD) / TB, TB, 0, stream>>>(x, proj_w, proj_b, pos, h);

  for (int l = 0; l < LL; ++l) {
    retention_kernel<<<BN, 256, 0, stream>>>(
        h, wqT + l * DD * DD, wkT + l * DD * DD, wvT + l * DD * DD,
        wq_b + l * DD, wk_b + l * DD, wv_b + l * DD,
        gn_w + l * DD, gn_b + l * DD, ln_w + l * DD, ln_b + l * DD,
        gamma + l, hret);
    diff_init_kernel<<<128, 256, 0, stream>>>(hret, dfT + l * DD * DD, conv_w + l * 3, cura, acc);
    diff_hop_kernel<<<128, 256, 0, stream>>>(adj, cura, rsum, conv_w + l * 3, 1, curb, acc);
    diff_hop_kernel<<<128, 256, 0, stream>>>(adj, curb, rsum, conv_w + l * 3, 2, cura, acc);
    diff_act_kernel<<<(BN * DD) / TB, TB, 0, stream>>>(acc, conv_b + l);
    fusion_kernel<<<BN, 256, 0, stream>>>(hret, acc, mlpT + l * 2 * DD * DD, mlp_b + l * DD, h);
  }
  head_kernel<<<BN, 64, 0, stream>>>(h, head1_w, head1_b, head2_w, head2_b, out);
}